// FFGNN_10960756539506
// MI455X (gfx1250) — compile-verified
//
#include <hip/hip_runtime.h>
#include <math.h>

typedef __attribute__((ext_vector_type(2))) float v2f;
typedef __attribute__((ext_vector_type(8))) float v8f;

#define TPB 256
#define LR_C 0.005f

// ---------------- degree / dinv ----------------
__global__ void k_init_deg(float* deg, int n) {
  int i = blockIdx.x * blockDim.x + threadIdx.x;
  if (i < n) deg[i] = 1.0f;  // self loop
}

__global__ void k_count_deg(const int* __restrict__ dst, float* deg, int e) {
  int i = blockIdx.x * blockDim.x + threadIdx.x;
  if (i < e) atomicAdd(&deg[dst[i]], 1.0f);
}

__global__ void k_deg2dinv(float* deg, int n) {
  int i = blockIdx.x * blockDim.x + threadIdx.x;
  if (i < n) {
    float d = deg[i];
    deg[i] = (d > 0.f) ? rsqrtf(d) : 0.f;
  }
}

// ---------------- u = dinv * (x @ W^T) via V_WMMA_F32_16X16X4_F32 ----------------
// One wave computes one 16(node)x16(out) tile, chaining 4 K=4 WMMAs, then scales
// each row by dinv[row] in the epilogue (u feeds both self-loop and edge messages).
// A[m][k] = x[node m][k]; B[k][j] = W[j][k]; C[m][j] = h tile.
__global__ void k_gemm_u(const float* __restrict__ x, const float* __restrict__ W,
                         const float* __restrict__ dinv, float* __restrict__ u,
                         int ntiles) {
  int wave = (blockIdx.x * blockDim.x + threadIdx.x) >> 5;
  if (wave >= ntiles) return;               // wave-uniform: EXEC stays all-ones
  int lane = threadIdx.x & 31;
  int half = lane >> 4;                     // 0: lanes 0-15, 1: lanes 16-31
  int l16  = lane & 15;
  size_t base = (size_t)wave * 16;

  const float2* xr = (const float2*)(x + (base + l16) * 16);   // row M=l16 of A
  const float2* wr = (const float2*)(W + (size_t)l16 * 16);    // col N=l16 of B
  v8f c = {};
#pragma unroll
  for (int kb = 0; kb < 4; ++kb) {          // K chunks of 4
    float2 av = xr[kb * 2 + half];          // A vgpr0/1 <-> K = 4kb+2*half, +1
    float2 bv = wr[kb * 2 + half];          // B vgpr0/1 <-> K rows {0,2}/{1,3} per half
    v2f a, b;
    a[0] = av.x; a[1] = av.y;
    b[0] = bv.x; b[1] = bv.y;
    c = __builtin_amdgcn_wmma_f32_16x16x4_f32(false, a, false, b, (short)0, c,
                                              false, false);
  }
#pragma unroll
  for (int i = 0; i < 8; ++i) {             // C vgpr i <-> row M = i + 8*half, col N = l16
    size_t row = base + (size_t)(i + half * 8);
    u[row * 16 + l16] = dinv[row] * c[i];
  }
}

// ---------------- edge scatter: acc[dst] += u[src] ------------------------------
__global__ void k_edge_scatter(const int* __restrict__ src, const int* __restrict__ dst,
                               const float* __restrict__ u, float* __restrict__ acc,
                               int e) {
  int i = blockIdx.x * blockDim.x + threadIdx.x;
  if (i >= e) return;
  int s = src[i];
  int d = dst[i];
  const float4* us = (const float4*)(u + (size_t)s * 16);
  float* ad = acc + (size_t)d * 16;
#pragma unroll
  for (int q = 0; q < 4; ++q) {
    float4 v = us[q];
    atomicAdd(ad + q * 4 + 0, v.x);
    atomicAdd(ad + q * 4 + 1, v.y);
    atomicAdd(ad + q * 4 + 2, v.z);
    atomicAdd(ad + q * 4 + 3, v.w);
  }
}

// ---- fused: z = relu(dinv*acc + b) in place; db += colsum(4z^2);
//      dW[k][j] += sum_i x[i][k] * 4*z[i][j]^2 via WMMA K-reduction.
// Chunk of 4 nodes x 16 cols = 64 elements = 32 lanes x 2: lane (half,l16) owns
// rows {i0+2*half, i0+2*half+1} at column l16 -> exactly the A/B WMMA layout.
__global__ void k_relu_dw_wmma(const float* __restrict__ x, float* __restrict__ zacc,
                               const float* __restrict__ dinv,
                               const float* __restrict__ bvec,
                               float* __restrict__ dW, float* __restrict__ dbL,
                               int nchunks) {
  __shared__ float sdb[16];
  for (int t = threadIdx.x; t < 16; t += blockDim.x) sdb[t] = 0.f;
  __syncthreads();
  int wave   = (blockIdx.x * blockDim.x + threadIdx.x) >> 5;
  int nwaves = (gridDim.x * blockDim.x) >> 5;
  int lane = threadIdx.x & 31;
  int half = lane >> 4;
  int l16  = lane & 15;
  float bcol = bvec[l16];
  float dbp = 0.f;
  v8f c = {};
  for (int ch = wave; ch < nchunks; ch += nwaves) {   // wave-uniform loop
    size_t i0   = (size_t)ch * 4;
    size_t row0 = i0 + 2 * half;
    size_t r0   = row0 * 16 + l16;
    float d0 = dinv[row0];
    float d1 = dinv[row0 + 1];
    float a0 = zacc[r0];
    float a1 = zacc[r0 + 16];
    float z0 = fmaxf(fmaf(d0, a0, bcol), 0.f);
    float z1 = fmaxf(fmaf(d1, a1, bcol), 0.f);
    zacc[r0]      = z0;                                // z written in place
    zacc[r0 + 16] = z1;
    float w0 = 4.f * z0 * z0;                          // wd = (2|z|)^2
    float w1 = 4.f * z1 * z1;
    dbp += w0 + w1;
    v2f a, b;
    a[0] = x[r0];                                      // A = x^T chunk
    a[1] = x[r0 + 16];
    b[0] = w0;                                         // B = wd chunk
    b[1] = w1;
    c = __builtin_amdgcn_wmma_f32_16x16x4_f32(false, a, false, b, (short)0, c,
                                              false, false);
  }
#pragma unroll
  for (int i = 0; i < 8; ++i)
    atomicAdd(&dW[(i + half * 8) * 16 + l16], c[i]);   // dW row M=k, col N=j
  atomicAdd(&sdb[l16], dbp);
  __syncthreads();
  if (threadIdx.x < 16) atomicAdd(&dbL[threadIdx.x], sdb[threadIdx.x]);
}

// ---------------- final: loss + Wc/bc writeout ----------------------------------
__global__ void k_writeout(const float* __restrict__ W1, const float* __restrict__ b1,
                           const float* __restrict__ dWa, const float* __restrict__ dba,
                           const int* __restrict__ positive, float* __restrict__ out) {
  int t = threadIdx.x;
  if (t == 0) {
    int pos = *positive;
    float agg = 0.f;
    for (int L = 0; L < 3; ++L) {
      float s = 0.f;
      for (int j = 0; j < 16; ++j) s += dba[L * 16 + j];
      float g = s * (1.f / 64.f);  // goodness = sum(z^2)/16 = sum(db)/64
      float loss;
      if (pos) {
        loss = (g > 10.f) ? 0.f : ((g < -10.f) ? -g : log1pf(expf(-g)));
      } else {
        loss = (g > 10.f) ? g : ((g < -10.f) ? 0.f : log1pf(expf(g)));
      }
      agg += loss;
    }
    out[0] = agg;
  }
  if (t < 256) {  // Wc[j][k] = W1[j][k] - LR * sum_L dW_L[k][j]
    int j = t >> 4, k = t & 15;
    float s = dWa[0 * 256 + k * 16 + j] + dWa[1 * 256 + k * 16 + j] +
              dWa[2 * 256 + k * 16 + j];
    out[1 + t] = W1[t] - LR_C * s;
  }
  if (t < 16) {
    out[257 + t] = b1[t] - LR_C * (dba[t] + dba[16 + t] + dba[32 + t]);
  }
}

extern "C" void kernel_launch(void* const* d_in, const int* in_sizes, int n_in,
                              void* d_out, int out_size, void* d_ws, size_t ws_size,
                              hipStream_t stream) {
  const float* x  = (const float*)d_in[0];
  const int*   ei = (const int*)d_in[1];
  const float* W1 = (const float*)d_in[2];
  const float* b1 = (const float*)d_in[3];
  const float* W2 = (const float*)d_in[4];
  const float* b2 = (const float*)d_in[5];
  const float* W3 = (const float*)d_in[6];
  const float* b3 = (const float*)d_in[7];
  const int* positive = (const int*)d_in[8];

  const int N = in_sizes[0] / 16;
  const int E = in_sizes[1] / 2;
  const int* src = ei;
  const int* dst = ei + E;

  float* ws   = (float*)d_ws;
  float* dinv = ws;                          // N
  float* U    = dinv + (size_t)N;            // N*16  (u = dinv * x@W^T)
  float* Z1   = U + (size_t)N * 16;          // N*16  (acc -> z, layers 1 & 3)
  float* Z2   = Z1 + (size_t)N * 16;         // N*16  (acc -> z, layer 2)
  float* dWa  = Z2 + (size_t)N * 16;         // 3*256
  float* dba  = dWa + 3 * 256;               // 3*16

  hipMemsetAsync(dWa, 0, (3 * 256 + 3 * 16) * sizeof(float), stream);

  int nb = (N + TPB - 1) / TPB;
  int eb = (E + TPB - 1) / TPB;
  k_init_deg<<<nb, TPB, 0, stream>>>(dinv, N);
  k_count_deg<<<eb, TPB, 0, stream>>>(dst, dinv, E);
  k_deg2dinv<<<nb, TPB, 0, stream>>>(dinv, N);

  const int ntiles  = N / 16;                        // N divisible by 16
  const int gemm_b  = (ntiles * 32 + TPB - 1) / TPB;
  const int nchunks = N / 4;

  const float* Ws[3] = {W1, W2, W3};
  const float* bs[3] = {b1, b2, b3};
  float* zbuf[3]     = {Z1, Z2, Z1};
  const float* xc = x;
  for (int L = 0; L < 3; ++L) {
    float* Z = zbuf[L];
    k_gemm_u<<<gemm_b, TPB, 0, stream>>>(xc, Ws[L], dinv, U, ntiles);
    // acc init = self-loop term: acc[i] = u[i]
    hipMemcpyAsync(Z, U, (size_t)N * 16 * sizeof(float),
                   hipMemcpyDeviceToDevice, stream);
    k_edge_scatter<<<eb, TPB, 0, stream>>>(src, dst, U, Z, E);
    k_relu_dw_wmma<<<512, TPB, 0, stream>>>(xc, Z, dinv, bs[L], dWa + L * 256,
                                            dba + L * 16, nchunks);
    xc = Z;
  }
  k_writeout<<<1, TPB, 0, stream>>>(W1, b1, dWa, dba, positive, (float*)d_out);
}